// SAGE_81011673137362
// MI455X (gfx1250) — compile-verified
//
#include <hip/hip_runtime.h>
#include <hip/hip_bf16.h>

// GraphSAGE (3-layer, mean aggregator) for MI455X / gfx1250.
// Memory-bound (~2.5GB traffic vs ~14 GFLOP): aggregation via f32 L2 atomics
// with b128 row gathers, GEMMs via fp32 WMMA (V_WMMA_F32_16X16X4_F32).

typedef __attribute__((ext_vector_type(2))) float v2f;
typedef __attribute__((ext_vector_type(8))) float v8f;

#define G_NODES 100000
#define G_EDGES 800000
#define G_IN    100
#define G_H     128
#define G_C     47

__global__ void fill_zero_kernel(float* __restrict__ p, long long n) {
  long long i = (long long)blockIdx.x * blockDim.x + threadIdx.x;
  if (i < n) p[i] = 0.0f;
}

__global__ void degree_kernel(const int* __restrict__ dst, float* __restrict__ deg, int E) {
  int e = blockIdx.x * blockDim.x + threadIdx.x;
  if (e < E) atomicAdd(&deg[dst[e]], 1.0f);
}

// 8 edges per 256-thread block: one wave32 per edge, no integer division.
// Lane j gathers float4 chunk j of h[src[e]] (coalesced b128 across the wave,
// rows are 16B-aligned: 100*4=400B, 128*4=512B) and scatter-adds into
// agg[dst[e]] (contiguous 512B atomic span per wave, L2-resident RMW).
template <int D>
__global__ __launch_bounds__(256)
void scatter_add_kernel(const float* __restrict__ h,
                        const int* __restrict__ src,
                        const int* __restrict__ dst,
                        float* __restrict__ agg, int E) {
  constexpr int D4 = D / 4;
  const int e = blockIdx.x * 8 + (threadIdx.x >> 5);
  const int lane = threadIdx.x & 31;
  if (e >= E) return;
  const int s = src[e];
  const int t = dst[e];
  const float4* __restrict__ hs = (const float4*)(h + (long long)s * D);
  float* __restrict__ ag = agg + (long long)t * D;
#pragma unroll
  for (int j = lane; j < D4; j += 32) {
    const float4 v = hs[j];
    atomicAdd(&ag[4 * j + 0], v.x);
    atomicAdd(&ag[4 * j + 1], v.y);
    atomicAdd(&ag[4 * j + 2], v.z);
    atomicAdd(&ag[4 * j + 3], v.w);
  }
}

// One wave per 16x16 output tile.
// out[m, n] = relu?( sum_k Hs[m,k]*Ws[k,n] + (Ag[m,k]/max(deg[m],1))*Wn[k,n] + bias[n] )
//
// WMMA f32 16x16x4 operand layout (ISA 7.12.2, wave32):
//   A 16x4 : lane L holds row M=L&15; VGPR0/1 = K = 2*(L>>4), 2*(L>>4)+1
//   B 4x16 : lane L holds col N=L&15; VGPR0/1 = K = 2*(L>>4), 2*(L>>4)+1
//   C/D    : 8 VGPRs, element r: M = r + 8*(L>>4), N = L&15
template <int K, int OUT, int RELU>
__global__ __launch_bounds__(32)
void sage_gemm_kernel(const float* __restrict__ Hs,    // [n_rows x K]
                      const float* __restrict__ Ag,    // [n_rows x K] (raw sums)
                      const float* __restrict__ deg,   // [n_rows]
                      const float* __restrict__ Ws,    // [K x OUT]
                      const float* __restrict__ Wn,    // [K x OUT]
                      const float* __restrict__ bias,  // [OUT]
                      float* __restrict__ out,         // [n_rows x OUT]
                      int n_rows) {
  const int lane = threadIdx.x & 31;
  const int la   = lane & 15;   // column within tile (B/C/D) and A-row
  const int half = lane >> 4;   // K sub-slot selector
  const int m0 = blockIdx.x * 16;
  const int n0 = blockIdx.y * 16;
  if (m0 >= n_rows) return;     // uniform per wave

  const int nn = n0 + la;
  // Branch-free column masking: keep EXEC all-1s through every WMMA.
  const int   nnc  = (nn < OUT) ? nn : (OUT - 1);
  const float nmsk = (nn < OUT) ? 1.0f : 0.0f;

  const int   mA  = m0 + la;                       // the A-row this lane owns
  const float inv = 1.0f / fmaxf(deg[mA], 1.0f);   // fold mean-normalization into A load

  v8f c;
  const float bv = bias[nnc] * nmsk;
#pragma unroll
  for (int r = 0; r < 8; ++r) c[r] = bv;

  const float* __restrict__ arow = Hs + (long long)mA * K + 2 * half;
  const float* __restrict__ grow = Ag + (long long)mA * K + 2 * half;

  // Self term: Hs @ Ws
#pragma unroll 4
  for (int k = 0; k < K; k += 4) {
    v2f a;
    a.x = arow[k];
    a.y = arow[k + 1];
    const int kr = k + 2 * half;
    v2f b;
    b.x = Ws[(long long)kr * OUT + nnc] * nmsk;
    b.y = Ws[(long long)(kr + 1) * OUT + nnc] * nmsk;
    c = __builtin_amdgcn_wmma_f32_16x16x4_f32(false, a, false, b, (short)0, c, false, false);
  }
  // Neighbor term: (Ag * invdeg) @ Wn
#pragma unroll 4
  for (int k = 0; k < K; k += 4) {
    v2f a;
    a.x = grow[k] * inv;
    a.y = grow[k + 1] * inv;
    const int kr = k + 2 * half;
    v2f b;
    b.x = Wn[(long long)kr * OUT + nnc] * nmsk;
    b.y = Wn[(long long)(kr + 1) * OUT + nnc] * nmsk;
    c = __builtin_amdgcn_wmma_f32_16x16x4_f32(false, a, false, b, (short)0, c, false, false);
  }

  if (nn < OUT) {
#pragma unroll
    for (int r = 0; r < 8; ++r) {
      const int m = m0 + r + 8 * half;
      float v = c[r];
      if (RELU) v = fmaxf(v, 0.0f);
      out[(long long)m * OUT + nn] = v;
    }
  }
}

static inline int div_up_ll(long long a, int b) { return (int)((a + b - 1) / b); }

extern "C" void kernel_launch(void* const* d_in, const int* in_sizes, int n_in,
                              void* d_out, int out_size, void* d_ws, size_t ws_size,
                              hipStream_t stream) {
  const float* x   = (const float*)d_in[0];
  const int*   src = (const int*)d_in[1];
  const int*   dst = (const int*)d_in[2];
  const float* Ws0 = (const float*)d_in[3];
  const float* Wn0 = (const float*)d_in[4];
  const float* b0  = (const float*)d_in[5];
  const float* Ws1 = (const float*)d_in[6];
  const float* Wn1 = (const float*)d_in[7];
  const float* b1  = (const float*)d_in[8];
  const float* Ws2 = (const float*)d_in[9];
  const float* Wn2 = (const float*)d_in[10];
  const float* b2  = (const float*)d_in[11];

  const int N = G_NODES, E = G_EDGES;

  // Workspace layout (fp32): deg[N] | agg[N*128] | h1[N*128] | h2[N*128]  (~154 MB)
  float* deg = (float*)d_ws;
  float* agg = deg + N;
  float* h1  = agg + (size_t)N * G_H;
  float* h2  = h1 + (size_t)N * G_H;

  const int edge_blocks = div_up_ll(E, 8);

  // Degrees (identical for all layers).
  fill_zero_kernel<<<div_up_ll(N, 256), 256, 0, stream>>>(deg, N);
  degree_kernel<<<div_up_ll(E, 256), 256, 0, stream>>>(dst, deg, E);

  // ---- Layer 0: x[100] -> h1[128], ReLU
  {
    const long long tot = (long long)N * G_IN;
    fill_zero_kernel<<<div_up_ll(tot, 256), 256, 0, stream>>>(agg, tot);
    scatter_add_kernel<G_IN><<<edge_blocks, 256, 0, stream>>>(x, src, dst, agg, E);
    dim3 g(N / 16, G_H / 16);
    sage_gemm_kernel<G_IN, G_H, 1><<<g, 32, 0, stream>>>(x, agg, deg, Ws0, Wn0, b0, h1, N);
  }
  // ---- Layer 1: h1[128] -> h2[128], ReLU
  {
    const long long tot = (long long)N * G_H;
    fill_zero_kernel<<<div_up_ll(tot, 256), 256, 0, stream>>>(agg, tot);
    scatter_add_kernel<G_H><<<edge_blocks, 256, 0, stream>>>(h1, src, dst, agg, E);
    dim3 g(N / 16, G_H / 16);
    sage_gemm_kernel<G_H, G_H, 1><<<g, 32, 0, stream>>>(h1, agg, deg, Ws1, Wn1, b1, h2, N);
  }
  // ---- Layer 2: h2[128] -> out[47], no ReLU
  {
    const long long tot = (long long)N * G_H;
    fill_zero_kernel<<<div_up_ll(tot, 256), 256, 0, stream>>>(agg, tot);
    scatter_add_kernel<G_H><<<edge_blocks, 256, 0, stream>>>(h2, src, dst, agg, E);
    dim3 g(N / 16, (G_C + 15) / 16);
    sage_gemm_kernel<G_H, G_C, 0><<<g, 32, 0, stream>>>(h2, agg, deg, Ws2, Wn2, b2, (float*)d_out, N);
  }
}